// GlobalFeatureBlock_Diffusion_88905823027691
// MI455X (gfx1250) — compile-verified
//
#include <hip/hip_runtime.h>

// ---------------------------------------------------------------------------
// Fully-fused GlobalFeatureBlock diffusion for MI455X (gfx1250).
//
// Roofline: 33.5M elems, ~10 GFLOP total vs >= 268 MB mandatory traffic at
// 23.3 TB/s HBM -> pure bandwidth problem (~12us floor). s0 (134MB) fits in
// the 192MB L2, so halo-redundant recompute across tiles is L2-cheap.
// Strategy: ONE kernel; per (b,y) stage a 42x42 (W,C) region (tile 32x32 +
// halo K=5, wrap-indexed to match jnp.roll), compute dwconv/BN/ReLU, Sobel,
// and pre-folded diffusion coefficients in LDS, run all K=5 leapfrog
// iterations in LDS (shrinking valid region), apply the output BN+ReLU at
// the single global store. No workspace, no inter-kernel HBM round trips.
// Async global->LDS (ASYNCcnt) is used for the s0 staging.
// ---------------------------------------------------------------------------

#define Bn  8
#define Hn  128
#define Wn  128
#define Cn  256
#define Kn  5
#define DTf 0.2f
#define EPSf 0.001f

#define TX 32            // tile extent along W
#define TC 32            // tile extent along C
#define RG 42            // region = tile + 2*K halo
#define SX 46            // staged s0 extent along W (region + 2 for conv)
#define SC 46            // staged s0 extent along C
#define GX 44            // staged g extent along W  (region + 1 for ux)
#define G1C 44           // staged g1 extent along C (region + 1 for vy)

// LDS layout (float offsets)
#define OFF_SIN  0                      // 3*46*46 = 6348
#define OFF_SG   6348                   // 44*42   = 1848
#define OFF_SG1  (OFF_SG  + 1848)       // 42*44   = 1848
#define OFF_P0   (OFF_SG1 + 1848)       // 7 coeff arrays of 42*42 = 1764
#define OFF_PH   (OFF_P0  + 1764)
#define OFF_PXM  (OFF_PH  + 1764)
#define OFF_PXP  (OFF_PXM + 1764)
#define OFF_PCM  (OFF_PXP + 1764)
#define OFF_PCP  (OFF_PCM + 1764)
#define OFF_F    (OFF_PCP + 1764)
#define OFF_HA   (OFF_F   + 1764)
#define OFF_HB   (OFF_HA  + 1764)
#define LDS_FLOATS (OFF_HB + 1764)      // 25920 floats = 103680 bytes

// Address-space typedefs matching the builtin prototype (clang reported
// the parameter type as '__device__ int *', i.e. AS1/AS3 int pointers).
typedef __attribute__((address_space(1))) int gas_int;   // global (AS1)
typedef __attribute__((address_space(3))) int las_int;   // LDS    (AS3)

// ---- async global -> LDS copy (CDNA5 ASYNCcnt datapath) -------------------
__device__ __forceinline__ void cp_async_f32(const float* gsrc, float* ldst) {
#if defined(__gfx1250__) && __has_builtin(__builtin_amdgcn_global_load_async_to_lds_b32)
  __builtin_amdgcn_global_load_async_to_lds_b32(
      (gas_int*)(unsigned long long)gsrc,
      (las_int*)(unsigned int)(unsigned long long)ldst,   // low 32b = LDS offset
      0, 0);
#elif defined(__gfx1250__)
  unsigned int loff = (unsigned int)(unsigned long long)ldst;
  unsigned long long gaddr = (unsigned long long)gsrc;
  asm volatile("global_load_async_to_lds_b32 %0, %1, off"
               :: "v"(loff), "v"(gaddr) : "memory");
#else
  *ldst = *gsrc;
#endif
}

__device__ __forceinline__ void wait_async_all() {
#if defined(__gfx1250__)
#if __has_builtin(__builtin_amdgcn_s_wait_asynccnt)
  __builtin_amdgcn_s_wait_asynccnt(0);
#else
  asm volatile("s_wait_asynccnt 0x0" ::: "memory");
#endif
#endif
}

__global__ __launch_bounds__(256) void fused_diffusion_kernel(
    const float* __restrict__ s0,
    const float* __restrict__ wg,  const float* __restrict__ wg1,
    const float* __restrict__ gga, const float* __restrict__ gbe,
    const float* __restrict__ gme, const float* __restrict__ gva,
    const float* __restrict__ g1ga,const float* __restrict__ g1be,
    const float* __restrict__ g1me,const float* __restrict__ g1va,
    const float* __restrict__ oga, const float* __restrict__ obe,
    const float* __restrict__ ome, const float* __restrict__ ova,
    float* __restrict__ out)
{
  extern __shared__ float lds[];
  float* sIn = lds + OFF_SIN;   // [3][SX][SC] rows y-1,y,y+1 (reflected rows)
  float* sG  = lds + OFF_SG;    // [GX][RG]
  float* sG1 = lds + OFF_SG1;   // [RG][G1C]
  float* cP0 = lds + OFF_P0;  float* cPh  = lds + OFF_PH;
  float* cPxm= lds + OFF_PXM; float* cPxp = lds + OFF_PXP;
  float* cPcm= lds + OFF_PCM; float* cPcp = lds + OFF_PCP;
  float* cF  = lds + OFF_F;
  float* hA  = lds + OFF_HA;  float* hB   = lds + OFF_HB;

  const int tid = threadIdx.x;
  const int x0  = blockIdx.x * TX;
  const int c0  = blockIdx.y * TC;
  const int bz  = blockIdx.z;            // b*H + y
  const int b   = bz >> 7;
  const int y   = bz & (Hn - 1);

  const int yT = (y == 0)      ? 1        : (y - 1);   // reflect (for Sobel)
  const int yB = (y == Hn - 1) ? (Hn - 2) : (y + 1);

  // ---- stage s0 region: 3 rows x 46 x 46, wrap-indexed (roll semantics) ----
  for (int i = tid; i < 3 * SX * SC; i += 256) {
    const int r  = i / (SX * SC);
    const int t  = i % (SX * SC);
    const int ii = t / SC;
    const int jj = t % SC;
    const int xg = (x0 - 7 + ii + Wn) & (Wn - 1);
    const int cg = (c0 - 7 + jj + Cn) & (Cn - 1);
    const int yr = (r == 0) ? yT : ((r == 1) ? y : yB);
    const float* gp = s0 + ((((size_t)b * Hn + yr) * Wn + xg) * Cn + cg);
    cp_async_f32(gp, &sIn[i]);
  }
  wait_async_all();
  __syncthreads();

  // ---- g = relu(bn(dwconv(s0, wg))) over x-extent 44, c-extent 42 ----------
  for (int i = tid; i < GX * RG; i += 256) {
    const int gxr = i / RG, rc = i % RG;
    const int xg = (x0 - 6 + gxr + Wn) & (Wn - 1);
    const int cg = (c0 - 5 + rc  + Cn) & (Cn - 1);
    float acc = 0.f;
#pragma unroll
    for (int dy = -1; dy <= 1; ++dy) {
      if ((unsigned)(y + dy) >= (unsigned)Hn) continue;        // zero 'SAME' pad
#pragma unroll
      for (int dx = -1; dx <= 1; ++dx) {
        if ((xg == 0 && dx < 0) || (xg == Wn - 1 && dx > 0)) continue;
        const float v = sIn[((dy + 1) * SX + (gxr + 1 + dx)) * SC + (rc + 2)];
        const float w = wg[((dy + 1) * 3 + (dx + 1)) * Cn + cg];
        acc = fmaf(v, w, acc);
      }
    }
    float bnv = (acc - gme[cg]) * rsqrtf(gva[cg] + EPSf) * gga[cg] + gbe[cg];
    sG[i] = bnv > 0.f ? bnv : 0.f;
  }

  // ---- g1 over x-extent 42, c-extent 44 ------------------------------------
  for (int i = tid; i < RG * G1C; i += 256) {
    const int rx = i / G1C, gcr = i % G1C;
    const int xg = (x0 - 5 + rx  + Wn) & (Wn - 1);
    const int cg = (c0 - 6 + gcr + Cn) & (Cn - 1);
    float acc = 0.f;
#pragma unroll
    for (int dy = -1; dy <= 1; ++dy) {
      if ((unsigned)(y + dy) >= (unsigned)Hn) continue;
#pragma unroll
      for (int dx = -1; dx <= 1; ++dx) {
        if ((xg == 0 && dx < 0) || (xg == Wn - 1 && dx > 0)) continue;
        const float v = sIn[((dy + 1) * SX + (rx + 2 + dx)) * SC + (gcr + 1)];
        const float w = wg1[((dy + 1) * 3 + (dx + 1)) * Cn + cg];
        acc = fmaf(v, w, acc);
      }
    }
    float bnv = (acc - g1me[cg]) * rsqrtf(g1va[cg] + EPSf) * g1ga[cg] + g1be[cg];
    sG1[i] = bnv > 0.f ? bnv : 0.f;
  }
  __syncthreads();

  // ---- Sobel + pre-folded diffusion coefficients over 42x42 region ---------
  for (int i = tid; i < RG * RG; i += 256) {
    const int rx = i / RG, rc = i % RG;
    const int xg = (x0 - 5 + rx + Wn) & (Wn - 1);
    const int ii = rx + 2, jj = rc + 2;
    const int dxm = (xg == 0)      ? +1 : -1;   // reflect step to x-1 tap
    const int dxp = (xg == Wn - 1) ? -1 : +1;   // reflect step to x+1 tap
    const float tm = sIn[(0 * SX + ii + dxm) * SC + jj];
    const float tc = sIn[(0 * SX + ii      ) * SC + jj];
    const float tp = sIn[(0 * SX + ii + dxp) * SC + jj];
    const float mm = sIn[(1 * SX + ii + dxm) * SC + jj];
    const float fs = sIn[(1 * SX + ii      ) * SC + jj];   // fsrc = s0 center
    const float mp = sIn[(1 * SX + ii + dxp) * SC + jj];
    const float bm = sIn[(2 * SX + ii + dxm) * SC + jj];
    const float bc = sIn[(2 * SX + ii      ) * SC + jj];
    const float bp = sIn[(2 * SX + ii + dxp) * SC + jj];
    const float gy = (bm + 2.f * bc + bp) - (tm + 2.f * tc + tp);
    const float gx = (tp + 2.f * mp + bp) - (tm + 2.f * mm + bm);
    const float Dxv = 1.f / (gy * gy * 0.25f + 1.f);       // LAM2 = 4
    const float Dyv = 1.f / (gx * gx * 0.25f + 1.f);
    const float g   = sG[(rx + 1) * RG + rc];
    const float gxm = sG[(rx    ) * RG + rc];
    const float gxp = sG[(rx + 2) * RG + rc];
    const float g1  = sG1[rx * G1C + rc + 1];
    const float g1m = sG1[rx * G1C + rc];
    const float g1p = sG1[rx * G1C + rc + 2];
    const float ux = 0.5f * (gxm - gxp);                   // roll(+/-1, axis W)
    const float vy = 0.5f * (g1m - g1p);                   // roll(+/-1, axis C)
    const float Ax = g * DTf, Ay = g1 * DTf;
    const float Bx = Dxv * DTf, By = Dyv * DTf;
    const float E  = (ux + vy) * DTf;
    const float Dd = 1.f / (1.f + 2.f * Bx + 2.f * By);
    const float c0v = 1.f - 2.f * Bx - 2.f * By;
    cP0[i]  = Dd * c0v;
    cPh[i]  = -2.f * Dd * E;
    cPxm[i] = Dd * (2.f * Bx - Ax);
    cPxp[i] = Dd * (2.f * Bx + Ax);
    cPcm[i] = Dd * (2.f * By - Ay);
    cPcp[i] = Dd * (2.f * By + Ay);
    cF[i]   = Dd * (2.f * DTf) * fs;
    hA[i] = fs;                                            // h = h0 = fsrc = s0
    hB[i] = fs;
  }
  __syncthreads();

  // ---- K=5 leapfrog iterations, shrinking valid region ---------------------
  float* cur = hA;    // h
  float* prev = hB;   // h0 (read only at center -> safe in-place target)
#pragma unroll
  for (int t = 0; t < Kn; ++t) {
    const int lo = t + 1;
    const int n  = RG - 2 * (t + 1);          // 40 - 2t
    for (int i = tid; i < n * n; i += 256) {
      const int rx = lo + i / n, rc = lo + i % n;
      const int idx = rx * RG + rc;
      const float hc = cur[idx];
      float nv = cP0[idx] * prev[idx] + cPh[idx] * hc
               + cPxm[idx] * cur[idx - RG] + cPxp[idx] * cur[idx + RG]
               + cPcm[idx] * cur[idx - 1]  + cPcp[idx] * cur[idx + 1]
               + cF[idx];
      prev[idx] = nv;
    }
    __syncthreads();
    float* tmp = cur; cur = prev; prev = tmp; // new h = written buf; h0 = old h
  }

  // ---- final BN + ReLU fused into the single global store ------------------
  for (int i = tid; i < TX * TC; i += 256) {
    const int tx = i / TC, tc2 = i % TC;
    const int xg = x0 + tx, cg = c0 + tc2;
    const float h = cur[(tx + Kn) * RG + (tc2 + Kn)];
    float v = (h - ome[cg]) * rsqrtf(ova[cg] + EPSf) * oga[cg] + obe[cg];
    out[(((size_t)b * Hn + y) * Wn + xg) * Cn + cg] = v > 0.f ? v : 0.f;
  }
}

extern "C" void kernel_launch(void* const* d_in, const int* in_sizes, int n_in,
                              void* d_out, int out_size, void* d_ws, size_t ws_size,
                              hipStream_t stream) {
  (void)in_sizes; (void)n_in; (void)d_ws; (void)ws_size; (void)out_size;
  const float* s0   = (const float*)d_in[0];
  const float* wg   = (const float*)d_in[1];
  const float* wg1  = (const float*)d_in[2];
  const float* gga  = (const float*)d_in[3];
  const float* gbe  = (const float*)d_in[4];
  const float* gme  = (const float*)d_in[5];
  const float* gva  = (const float*)d_in[6];
  const float* g1ga = (const float*)d_in[7];
  const float* g1be = (const float*)d_in[8];
  const float* g1me = (const float*)d_in[9];
  const float* g1va = (const float*)d_in[10];
  const float* oga  = (const float*)d_in[11];
  const float* obe  = (const float*)d_in[12];
  const float* ome  = (const float*)d_in[13];
  const float* ova  = (const float*)d_in[14];
  float* out = (float*)d_out;

  const size_t shmem = (size_t)LDS_FLOATS * sizeof(float);   // ~104 KB of 320 KB/WGP
  (void)hipFuncSetAttribute(reinterpret_cast<const void*>(&fused_diffusion_kernel),
                            hipFuncAttributeMaxDynamicSharedMemorySize, (int)shmem);

  dim3 grid(Wn / TX, Cn / TC, Bn * Hn);   // 4 x 8 x 1024 blocks
  fused_diffusion_kernel<<<grid, 256, shmem, stream>>>(
      s0, wg, wg1, gga, gbe, gme, gva,
      g1ga, g1be, g1me, g1va, oga, obe, ome, ova, out);
}